// loss_generator_31482110279761
// MI455X (gfx1250) — compile-verified
//
#include <hip/hip_runtime.h>
#include <cstdint>
#include <cstddef>

// Problem geometry
#define T_FRAMES 402
#define H_IN     256
#define W_IN     256
#define T_OUT    400
#define H_OUT    252
#define W_OUT    252

// Tile geometry: 256 threads = 64 x 4 outputs, halo 2 on each side.
#define TX    64
#define TY    4
#define SLDS  68          // (TX + 4) floats per LDS row
#define TROWS 8           // (TY + 4) rows

// Stencil constants (computed in double, stored as fp32)
constexpr double DXd   = 20.0 / 128.0;          // 0.15625
constexpr double DTd   = 10.0 / 200.0;          // 0.05
constexpr float  L1    = (float)((4.0 / 3.0)  / (DXd * DXd));
constexpr float  L2c   = (float)((1.0 / 12.0) / (DXd * DXd));
constexpr float  L0    = (float)(5.0          / (DXd * DXd));
constexpr float  K1    = (float)((8.0 / 12.0) / DXd);
constexpr float  K2    = (float)((1.0 / 12.0) / DXd);
constexpr float  I2DT  = (float)(1.0 / (2.0 * DTd));   // 10.0
constexpr float  INVR  = (float)(1.0 / 200.0);         // 0.005

#if defined(__AMDGCN__) && \
    __has_builtin(__builtin_amdgcn_global_load_async_to_lds_b128) && \
    __has_builtin(__builtin_amdgcn_s_wait_asynccnt)
#define HAVE_ASYNC_LDS 1
#else
#define HAVE_ASYNC_LDS 0
#endif

#if HAVE_ASYNC_LDS
// Builtin signature (from hipcc diagnostic): first param is
// `int __vector(4)` pointer in AS(1) (global), second in AS(3) (LDS).
typedef int v4i __attribute__((vector_size(16)));
typedef __attribute__((address_space(1))) v4i global_v4i;
typedef __attribute__((address_space(3))) v4i lds_v4i;

__device__ __forceinline__ void async_copy_b128(const float* g, float* l) {
    // Per-lane: copies 16 bytes from the global address g to the LDS address l, tracked on ASYNCcnt.
    // Flat addresses of global memory are numerically identical to AS1
    // addresses on AMDGCN, so the integer round-trip is safe for the global
    // side; the LDS side uses a real addrspacecast (generic -> AS3).
    __builtin_amdgcn_global_load_async_to_lds_b128(
        (global_v4i*)(uintptr_t)g, (lds_v4i*)l, /*offset=*/0, /*cpol=*/0);
}
#endif

__global__ __launch_bounds__(256) void burgers_residual_kernel(
    const float* __restrict__ in,   // (402, 2, 256, 256) fp32
    float* __restrict__ out)        // f_u (400,252,252) then f_v (400,252,252)
{
    __shared__ __align__(16) float su[TROWS * SLDS];
    __shared__ __align__(16) float sv[TROWS * SLDS];

    const int tid = threadIdx.x;
    const int x0  = blockIdx.x * TX;     // input col base of halo tile
    const int y0  = blockIdx.y * TY;     // input row base of halo tile
    const int ot  = blockIdx.z;          // output frame, uses input frames ot..ot+2
    const int m   = ot + 1;              // spatial frame

    const size_t FRAME = (size_t)H_IN * W_IN;
    const size_t base_u = (size_t)(m * 2 + 0) * FRAME;
    const size_t base_v = (size_t)(m * 2 + 1) * FRAME;

    // ---- Stage haloed tile of frame m (u and v) into LDS ----
#if HAVE_ASYNC_LDS
    if (tid < TROWS * (SLDS / 4)) {      // 8 rows x 17 sixteen-byte chunks = 136 lanes
        const int row   = tid / (SLDS / 4);
        const int chunk = (tid % (SLDS / 4)) * 4;       // float offset in row
        const size_t goff = (size_t)(y0 + row) * W_IN + (size_t)(x0 + chunk);
        async_copy_b128(in + base_u + goff, &su[row * SLDS + chunk]);
        async_copy_b128(in + base_v + goff, &sv[row * SLDS + chunk]);
    }
#else
    for (int l = tid; l < TROWS * SLDS; l += 256) {
        const int row = l / SLDS;
        int gx = x0 + (l - row * SLDS);
        if (gx > W_IN - 1) gx = W_IN - 1;               // halo clamp (unused lanes)
        const size_t goff = (size_t)(y0 + row) * W_IN + (size_t)gx;
        su[l] = in[base_u + goff];
        sv[l] = in[base_v + goff];
    }
#endif

    // ---- While the DMA is in flight: center-pixel loads for the time derivative ----
    const int lx = tid & (TX - 1);
    const int ly = tid >> 6;             // 0..3
    const int ox = x0 + lx;
    const int oy = y0 + ly;              // always < 252 (63 * 4 = 252 exactly)
    const bool valid = (ox < W_OUT);
    const int cy = oy + 2;
    int cx = ox + 2;
    if (cx > W_IN - 1) cx = W_IN - 1;    // clamp for the discarded right-edge lanes
    const size_t cpix = (size_t)cy * W_IN + (size_t)cx;

    const float u0 = in[(size_t)(ot * 2 + 0) * FRAME + cpix];
    const float v0 = in[(size_t)(ot * 2 + 1) * FRAME + cpix];
    const float u2 = in[(size_t)((ot + 2) * 2 + 0) * FRAME + cpix];
    const float v2 = in[(size_t)((ot + 2) * 2 + 1) * FRAME + cpix];

    // Pull the next time frame toward L2 (global_prefetch path; speculative).
    if (m + 1 < T_FRAMES)
        __builtin_prefetch(in + (size_t)((m + 1) * 2) * FRAME + cpix, 0, 1);

#if HAVE_ASYNC_LDS
    __builtin_amdgcn_s_wait_asynccnt(0);  // drain this wave's async copies
#endif
    __syncthreads();                      // make all waves' LDS writes visible

    // ---- Cross-stencil from LDS ----
    const int c0 = (ly + 2) * SLDS + (lx + 2);

    const float uC  = su[c0];
    const float uN1 = su[c0 - SLDS],     uS1 = su[c0 + SLDS];
    const float uN2 = su[c0 - 2 * SLDS], uS2 = su[c0 + 2 * SLDS];
    const float uW1 = su[c0 - 1],        uE1 = su[c0 + 1];
    const float uW2 = su[c0 - 2],        uE2 = su[c0 + 2];

    const float vC  = sv[c0];
    const float vN1 = sv[c0 - SLDS],     vS1 = sv[c0 + SLDS];
    const float vN2 = sv[c0 - 2 * SLDS], vS2 = sv[c0 + 2 * SLDS];
    const float vW1 = sv[c0 - 1],        vE1 = sv[c0 + 1];
    const float vW2 = sv[c0 - 2],        vE2 = sv[c0 + 2];

    const float lap_u = L1 * ((uN1 + uS1) + (uW1 + uE1))
                      - L2c * ((uN2 + uS2) + (uW2 + uE2))
                      - L0 * uC;
    const float lap_v = L1 * ((vN1 + vS1) + (vW1 + vE1))
                      - L2c * ((vN2 + vS2) + (vW2 + vE2))
                      - L0 * vC;

    // PART_X varies along H (u_x), PART_Y along W (u_y), 4th-order central.
    const float uXh = K1 * (uS1 - uN1) - K2 * (uS2 - uN2);
    const float uYw = K1 * (uE1 - uW1) - K2 * (uE2 - uW2);
    const float vXh = K1 * (vS1 - vN1) - K2 * (vS2 - vN2);
    const float vYw = K1 * (vE1 - vW1) - K2 * (vE2 - vW2);

    const float fu = (u2 - u0) * I2DT + uC * uXh + vC * uYw - INVR * lap_u;
    const float fv = (v2 - v0) * I2DT + uC * vXh + vC * vYw - INVR * lap_v;

    if (valid) {
        const size_t o = ((size_t)ot * H_OUT + (size_t)oy) * W_OUT + (size_t)ox;
        // Non-temporal: keep the 203 MB output stream from evicting the
        // input frames.
        __builtin_nontemporal_store(fu, &out[o]);
        __builtin_nontemporal_store(fv, &out[(size_t)T_OUT * H_OUT * W_OUT + o]);
    }
}

extern "C" void kernel_launch(void* const* d_in, const int* in_sizes, int n_in,
                              void* d_out, int out_size, void* d_ws, size_t ws_size,
                              hipStream_t stream) {
    (void)in_sizes; (void)n_in; (void)d_ws; (void)ws_size; (void)out_size;
    const float* in = (const float*)d_in[0];
    float* out = (float*)d_out;

    dim3 block(256, 1, 1);
    dim3 grid((W_OUT + TX - 1) / TX,   // 4
              (H_OUT + TY - 1) / TY,   // 63
              T_OUT);                  // 400
    burgers_residual_kernel<<<grid, block, 0, stream>>>(in, out);
}